// RoIAlignDenseAdaMax_73349451481355
// MI455X (gfx1250) — compile-verified
//
#include <hip/hip_runtime.h>
#include <cstdint>

typedef __attribute__((ext_vector_type(2))) float v2f;
typedef __attribute__((ext_vector_type(8))) float v8f;

#define C_TOT 256
#define Hf 100
#define Wf 100
#define GS 11            // dense grid 11x11
#define PO 5             // pooled output 5x5
#define SCALE 0.0625f
#define TILE 16          // staged tile is 16x16 per channel
#define CH_PER_ROUND 32
#define ROUNDS 8
#define PLANE 260        // padded per-channel plane stride (floats): 16*16 + 4
#define POOL_N (CH_PER_ROUND * PO * PO)   // 800 pooled outputs per round

__launch_bounds__(256)
__global__ void roi_align_pool_kernel(const float* __restrict__ feats,
                                      const float* __restrict__ rois,
                                      float* __restrict__ out)
{
    __shared__ float sKx[16 * 17];                     // x-interp weights, row-padded
    __shared__ float sTile[2][CH_PER_ROUND * PLANE];   // double-buffered channel planes
    __shared__ float sDns[2 * GS * 16 * 16];           // dense grid: 2 chunks x 11 x 16gx x 16c
    __shared__ int   sY0[GS];
    __shared__ int   sY1[GS];
    __shared__ float sWy[GS];

    const int k    = blockIdx.x;
    const int t    = threadIdx.x;
    const int lane = t & 31;
    const int wave = t >> 5;

    // ---- per-ROI scalars, computed redundantly by every thread ----
    const float r0  = rois[k * 5 + 0];
    const float x1s = rois[k * 5 + 1] * SCALE;
    const float y1s = rois[k * 5 + 2] * SCALE;
    const float x2s = rois[k * 5 + 3] * SCALE;
    const float y2s = rois[k * 5 + 4] * SCALE;
    const int   b   = (int)r0;

    int x0i0 = (int)floorf(x1s); x0i0 = min(max(x0i0, 0), Wf - 1);
    int y0i0 = (int)floorf(y1s); y0i0 = min(max(y0i0, 0), Hf - 1);
    int xbase = min(x0i0, Wf - TILE) & ~1;   // even in [0,84]: b64-aligned, covers all samples
    int ybase = min(y0i0, Hf - TILE);        // in [0,84]
    if (xbase < 0) xbase = 0;
    if (ybase < 0) ybase = 0;

    // ---- grid coords + Kx weight matrix (threads 0..15) ----
    if (t < GS) {
        const int   g  = t;
        const float tc = (float)g / 10.0f;     // arange(11)/(Hs-1)
        float ys  = y1s + (y2s - y1s) * tc;
        float y0f = floorf(ys);
        int   yi  = (int)y0f;
        sY0[g] = min(max(yi, 0), Hf - 1) - ybase;
        sY1[g] = min(max(yi + 1, 0), Hf - 1) - ybase;
        sWy[g] = ys - y0f;
        float xs  = x1s + (x2s - x1s) * tc;
        float x0f = floorf(xs);
        float wx  = xs - x0f;
        int   xi  = (int)x0f;
        int x0c = min(max(xi, 0), Wf - 1);
        int x1c = min(max(xi + 1, 0), Wf - 1);
        float* row = &sKx[g * 17];
        #pragma unroll
        for (int j = 0; j < 17; ++j) row[j] = 0.0f;
        row[x0c - xbase] += 1.0f - wx;        // clip-duplicates sum = reference 4-tap form
        row[x1c - xbase] += wx;
    } else if (t < 16) {
        float* row = &sKx[t * 17];             // zero unused A rows gx=11..15
        #pragma unroll
        for (int j = 0; j < 17; ++j) row[j] = 0.0f;
    }

    // ---- round-invariant pool index decomposition (hoisted out of round loop) ----
    int pcc[4], poy[4], pox[4];
    bool pval[4];
    #pragma unroll
    for (int i = 0; i < 4; ++i) {
        int o = t + 256 * i;
        pval[i] = (o < POOL_N);
        int cc = o / (PO * PO);
        int p  = o % (PO * PO);
        pcc[i] = cc;
        poy[i] = p / PO;
        pox[i] = p % PO;
    }

    const int      foffs_base = b * (C_TOT * Hf * Wf);
    const uint64_t fbase_sgpr = (uint64_t)(uintptr_t)feats;

    // ---- async stage of one 32-channel round into a tile buffer ----
    auto issue_stage = [&](int round, int buf) {
        const int ch0 = round * CH_PER_ROUND;
        #pragma unroll
        for (int i = 0; i < 16; ++i) {
            int q   = i * 256 + t;     // 4096 b64 transfers: 32ch * 16 rows * 8 pairs
            int ch  = q >> 7;
            int rem = q & 127;
            int row = rem >> 3;
            int grp = rem & 7;
            uint32_t lds_off = (uint32_t)(uintptr_t)&sTile[buf][ch * PLANE + row * 16 + grp * 2];
            uint32_t goff = (uint32_t)((foffs_base + (ch0 + ch) * (Hf * Wf)
                                        + (ybase + row) * Wf + xbase + grp * 2) * 4);
            asm volatile("global_load_async_to_lds_b64 %0, %1, %2"
                         :: "v"(lds_off), "v"(goff), "s"(fbase_sgpr)
                         : "memory");
        }
    };

    issue_stage(0, 0);    // prime the pipeline

    for (int round = 0; round < ROUNDS; ++round) {
        const int ch0 = round * CH_PER_ROUND;
        const int buf = round & 1;

        // prefetch next round into the other buffer, then wait for current buffer only
        if (round + 1 < ROUNDS) {
            issue_stage(round + 1, buf ^ 1);
            asm volatile("s_wait_asynccnt 16" ::: "memory");  // in-order: first 16 (cur buf) done
        } else {
            asm volatile("s_wait_asynccnt 0" ::: "memory");
        }
        __syncthreads();   // tile (+ round-0 coords/Kx) visible to all waves

        // ---- per-wave: 16x16x16 f32 GEMM  D[gx,c] = Kx * Ry  via 4x WMMA K=4 ----
        const int    chunk = wave & 1;     // which 16-channel half of this round
        const int    q4    = wave >> 1;    // gy strip
        const int    m     = lane & 15;    // M (gx) for A, N (c) for B/D
        const int    hi    = lane >> 4;
        const float* tbase = &sTile[buf][(chunk * 16 + m) * PLANE];

        for (int gy = q4; gy < GS; gy += 4) {
            const int   y0l = sY0[gy];
            const int   y1l = sY1[gy];
            const float wy  = sWy[gy];
            const float wy0 = 1.0f - wy;

            v8f acc = {};
            #pragma unroll
            for (int j = 0; j < 4; ++j) {
                const int kk = 4 * j + 2 * hi;    // K index per ISA 16x4 f32 A layout
                v2f a, bm;
                a.x = sKx[m * 17 + kk];
                a.y = sKx[m * 17 + kk + 1];
                float t00 = tbase[y0l * 16 + kk];
                float t01 = tbase[y0l * 16 + kk + 1];
                float t10 = tbase[y1l * 16 + kk];
                float t11 = tbase[y1l * 16 + kk + 1];
                bm.x = wy0 * t00 + wy * t10;      // y-interp feeds B
                bm.y = wy0 * t01 + wy * t11;
                acc = __builtin_amdgcn_wmma_f32_16x16x4_f32(
                          false, a, false, bm, (short)0, acc, false, false);
            }
            float* drow = &sDns[(chunk * GS + gy) * 256];
            #pragma unroll
            for (int r = 0; r < 8; ++r)
                drow[(r + 8 * hi) * 16 + m] = acc[r];
        }
        __syncthreads();   // dense grid complete across waves

        // ---- 3x3 stride-2 max pool over 11x11 -> 5x5, store ----
        #pragma unroll
        for (int i = 0; i < 4; ++i) {
            if (!pval[i]) continue;
            const int cc = pcc[i], oy = poy[i], ox = pox[i];
            const int chunk2 = cc >> 4, cl = cc & 15;
            const float* dbase = &sDns[chunk2 * GS * 256];
            float mx = -__builtin_huge_valf();
            #pragma unroll
            for (int dy = 0; dy < 3; ++dy)
                #pragma unroll
                for (int dx = 0; dx < 3; ++dx) {
                    float v = dbase[((2 * oy + dy) * 16 + (2 * ox + dx)) * 16 + cl];
                    mx = fmaxf(mx, v);
                }
            out[((k * C_TOT + ch0 + cc) * PO + oy) * PO + ox] = mx;
        }
        __syncthreads();   // protect dense buffer reuse next round
    }
}

extern "C" void kernel_launch(void* const* d_in, const int* in_sizes, int n_in,
                              void* d_out, int out_size, void* d_ws, size_t ws_size,
                              hipStream_t stream)
{
    const float* feats = (const float*)d_in[0];
    const float* rois  = (const float*)d_in[1];
    float* out = (float*)d_out;
    const int K = in_sizes[1] / 5;   // 2048 ROIs
    roi_align_pool_kernel<<<K, 256, 0, stream>>>(feats, rois, out);
}